// RGCN_47236050321804
// MI455X (gfx1250) — compile-verified
//
#include <hip/hip_runtime.h>
#include <hip/hip_bf16.h>

typedef __attribute__((ext_vector_type(2))) float v2f;
typedef __attribute__((ext_vector_type(8))) float v8f;

#define DIM 128

// ---------------- utility kernels ----------------

__global__ void rgcn_zero_kernel(float* __restrict__ p, long long n) {
    long long t = (long long)blockIdx.x * blockDim.x + threadIdx.x;
    if (t < n) p[t] = 0.0f;
}

__global__ void rgcn_degree_kernel(const int* __restrict__ idx,
                                   float* __restrict__ deg, int n) {
    int t = blockIdx.x * blockDim.x + threadIdx.x;
    if (t < n) atomicAdd(deg + idx[t], 1.0f);
}

__global__ void rgcn_rsqrt_clamp_kernel(float* __restrict__ p, int n) {
    int t = blockIdx.x * blockDim.x + threadIdx.x;
    if (t < n) {
        float d = p[t];
        d = d < 1.0f ? 1.0f : d;
        p[t] = 1.0f / sqrtf(d);
    }
}

// Pre-swizzle a layer's (3,128,128) W block into WMMA B-fragment order:
// frag = kk*8 + tile (kk = K/4 step, tile = N/16); per lane a contiguous v2f.
// Makes every B fragment one coalesced global_load_b64 at uniform_base+lane*8.
__global__ void rgcn_wswizzle_kernel(const float* __restrict__ W,   // (3,128,128)
                                     float* __restrict__ Wsw) {     // (3,8192) v2f
    int t = blockIdx.x * blockDim.x + threadIdx.x;     // rel*8192 + frag*32 + lane
    if (t >= 3 * 256 * 32) return;
    int lane = t & 31;
    int frag = (t >> 5) & 255;
    int rel  = t >> 13;
    int kk   = frag >> 3;
    int tile = frag & 7;
    int half = lane >> 4, l16 = lane & 15;
    int ks  = kk * 4 + half * 2;
    int col = tile * 16 + l16;
    const float* Wr = W + rel * DIM * DIM;
    float2 v;
    v.x = Wr[ks * DIM + col];
    v.y = Wr[(ks + 1) * DIM + col];
    ((float2*)Wsw)[t] = v;
}

// ---------------- edge aggregation ----------------
// One wave32 per edge; lane c handles float4 chunk c of the 128-float row.
__global__ __launch_bounds__(256)
void rgcn_aggregate_kernel(const float* __restrict__ feat,
                           const int* __restrict__ src,
                           const int* __restrict__ dst,
                           const float* __restrict__ rs_out,
                           float* __restrict__ agg, int n_edges) {
    long long t = (long long)blockIdx.x * blockDim.x + threadIdx.x;
    int e = (int)(t >> 5);
    int c = (int)(t & 31);
    if (e >= n_edges) return;
    int s = src[e];
    int d = dst[e];
    float sc = rs_out[s];
    const float4* fp = (const float4*)(feat + (size_t)s * DIM);
    float4 v = fp[c];
    float* out = agg + (size_t)d * DIM + c * 4;
    atomicAdd(out + 0, v.x * sc);
    atomicAdd(out + 1, v.y * sc);
    atomicAdd(out + 2, v.z * sc);
    atomicAdd(out + 3, v.w * sc);
}

// ---------------- WMMA GEMM ----------------
// C[n_rows x 128] = maybe_relu( (A0 .* s0) @ W0 + b0  [ + (A1 .* s1) @ W1 + b1 ] )
// 8 waves/block; each wave computes 32 rows x 128 cols (2 M-tiles x 8 N-tiles,
// 16 v8f accumulators) so every B fragment load feeds two v_wmma_f32_16x16x4_f32.
__global__ __launch_bounds__(256)
void rgcn_gemm_kernel(const float* __restrict__ A0, const float* __restrict__ s0,
                      const float* __restrict__ Wsw0, const float* __restrict__ b0,
                      const float* __restrict__ A1, const float* __restrict__ s1,
                      const float* __restrict__ Wsw1, const float* __restrict__ b1,
                      float* __restrict__ C, int n_rows, int do_relu) {
    const int lane = threadIdx.x & 31;
    const int wave = threadIdx.x >> 5;
    const int half = lane >> 4;     // 0: K pair {k,k+1}, 1: {k+2,k+3}
    const int l16  = lane & 15;
    const int row_base = blockIdx.x * 256 + wave * 32;

    int ar0 = row_base + l16;
    int ar1 = row_base + 16 + l16;
    int ar0c = ar0 < n_rows ? ar0 : (n_rows - 1);
    int ar1c = ar1 < n_rows ? ar1 : (n_rows - 1);

    v8f acc[2][8];
#pragma unroll
    for (int t = 0; t < 8; ++t) {
        float bv = b0[t * 16 + l16];
        if (b1) bv += b1[t * 16 + l16];
#pragma unroll
        for (int r = 0; r < 8; ++r) { acc[0][t][r] = bv; acc[1][t][r] = bv; }
    }

    const int npass = A1 ? 2 : 1;
    for (int pass = 0; pass < npass; ++pass) {           // uniform branch
        const float* A   = pass ? A1 : A0;
        const float* S   = pass ? s1 : s0;
        const float* Wsw = pass ? Wsw1 : Wsw0;
        const float sc0 = S[ar0c];
        const float sc1 = S[ar1c];
        const v2f* a0p = (const v2f*)(A + (size_t)ar0c * DIM + half * 2);
        const v2f* a1p = (const v2f*)(A + (size_t)ar1c * DIM + half * 2);
        const v2f* wp  = (const v2f*)Wsw + lane;         // +frag*32 below
#pragma unroll 2
        for (int kk = 0; kk < 32; ++kk) {
            v2f a0 = __builtin_nontemporal_load(&a0p[kk * 2]);   // streamed once
            v2f a1 = __builtin_nontemporal_load(&a1p[kk * 2]);
            a0.x *= sc0; a0.y *= sc0;
            a1.x *= sc1; a1.y *= sc1;
#pragma unroll
            for (int t = 0; t < 8; ++t) {
                v2f b = wp[(kk * 8 + t) * 32];           // coalesced b64, cached
                acc[0][t] = __builtin_amdgcn_wmma_f32_16x16x4_f32(
                    false, a0, false, b, (short)0, acc[0][t], false, false);
                acc[1][t] = __builtin_amdgcn_wmma_f32_16x16x4_f32(
                    false, a1, false, b, (short)0, acc[1][t], false, false);
            }
        }
    }

    // store: acc[m][t] VGPR r <-> row row_base + m*16 + half*8 + r, col t*16+l16
#pragma unroll
    for (int m = 0; m < 2; ++m) {
#pragma unroll
        for (int t = 0; t < 8; ++t) {
            const int col = t * 16 + l16;
#pragma unroll
            for (int r = 0; r < 8; ++r) {
                const int row = row_base + m * 16 + half * 8 + r;
                float v = acc[m][t][r];
                if (do_relu) v = fmaxf(v, 0.0f);
                if (row < n_rows) C[(size_t)row * DIM + col] = v;
            }
        }
    }
}

// ---------------- orchestration ----------------

static inline int cdiv(long long a, long long b) { return (int)((a + b - 1) / b); }

extern "C" void kernel_launch(void* const* d_in, const int* in_sizes, int n_in,
                              void* d_out, int out_size, void* d_ws, size_t ws_size,
                              hipStream_t stream) {
    const float* drug_feats = (const float*)d_in[0];
    const float* prot_feats = (const float*)d_in[1];
    const int* ddi_src = (const int*)d_in[2];
    const int* ddi_dst = (const int*)d_in[3];
    const int* pdi_src = (const int*)d_in[4];
    const int* pdi_dst = (const int*)d_in[5];
    const int* ppi_src = (const int*)d_in[6];
    const int* ppi_dst = (const int*)d_in[7];
    const float* Wall[3] = { (const float*)d_in[8], (const float*)d_in[10], (const float*)d_in[12] };
    const float* ball[3] = { (const float*)d_in[9], (const float*)d_in[11], (const float*)d_in[13] };
    float* out = (float*)d_out;

    const int ND = in_sizes[0] / DIM;
    const int NP = in_sizes[1] / DIM;
    const int E_DDI = in_sizes[2];
    const int E_PDI = in_sizes[4];
    const int E_PPI = in_sizes[6];

    // workspace layout (floats)
    float* ws = (float*)d_ws;
    float* rs_ddi_out = ws;                    // ND
    float* rs_ddi_in  = rs_ddi_out + ND;       // ND
    float* rs_pdi_out = rs_ddi_in + ND;        // NP
    float* rs_pdi_in  = rs_pdi_out + NP;       // ND
    float* rs_ppi_out = rs_pdi_in + ND;        // NP
    float* rs_ppi_in  = rs_ppi_out + NP;       // NP
    const long long n_deg = 3LL * ND + 3LL * NP;
    float* agg_dd = rs_ppi_in + NP;            // ND*DIM
    float* agg_dp = agg_dd + (size_t)ND * DIM; // ND*DIM
    float* agg_pp = agg_dp + (size_t)ND * DIM; // NP*DIM
    float* hd_buf = agg_pp + (size_t)NP * DIM; // ND*DIM
    float* hp_buf = hd_buf + (size_t)ND * DIM; // NP*DIM
    float* wsw    = hp_buf + (size_t)NP * DIM; // 9 * 128*128 swizzled weights
    const long long n_agg = (2LL * ND + NP) * DIM;

    const int TB = 256;

    // swizzle all 9 weight matrices into B-fragment order (3 relations/layer)
    for (int layer = 0; layer < 3; ++layer)
        rgcn_wswizzle_kernel<<<cdiv(3 * 256 * 32, TB), TB, 0, stream>>>(
            Wall[layer], wsw + (size_t)layer * 3 * DIM * DIM);

    // degrees -> 1/sqrt(clamp(deg,1))
    rgcn_zero_kernel<<<cdiv(n_deg, TB), TB, 0, stream>>>(rs_ddi_out, n_deg);
    rgcn_degree_kernel<<<cdiv(E_DDI, TB), TB, 0, stream>>>(ddi_src, rs_ddi_out, E_DDI);
    rgcn_degree_kernel<<<cdiv(E_DDI, TB), TB, 0, stream>>>(ddi_dst, rs_ddi_in, E_DDI);
    rgcn_degree_kernel<<<cdiv(E_PDI, TB), TB, 0, stream>>>(pdi_src, rs_pdi_out, E_PDI);
    rgcn_degree_kernel<<<cdiv(E_PDI, TB), TB, 0, stream>>>(pdi_dst, rs_pdi_in, E_PDI);
    rgcn_degree_kernel<<<cdiv(E_PPI, TB), TB, 0, stream>>>(ppi_src, rs_ppi_out, E_PPI);
    rgcn_degree_kernel<<<cdiv(E_PPI, TB), TB, 0, stream>>>(ppi_dst, rs_ppi_in, E_PPI);
    rgcn_rsqrt_clamp_kernel<<<cdiv(n_deg, TB), TB, 0, stream>>>(rs_ddi_out, (int)n_deg);

    const float* hd = drug_feats;
    const float* hp = prot_feats;

    for (int layer = 0; layer < 3; ++layer) {
        const float* Wsw = wsw + (size_t)layer * 3 * DIM * DIM;
        const float* b = ball[layer];   // (3,128)
        const int relu = (layer < 2) ? 1 : 0;
        float* hd_next = (layer == 2) ? out : hd_buf;
        float* hp_next = (layer == 2) ? (out + (size_t)ND * DIM) : hp_buf;

        // zero all three agg buffers (contiguous)
        rgcn_zero_kernel<<<cdiv(n_agg, TB), TB, 0, stream>>>(agg_dd, n_agg);

        // edge scatter-adds
        rgcn_aggregate_kernel<<<cdiv((long long)E_DDI * 32, TB), TB, 0, stream>>>(
            hd, ddi_src, ddi_dst, rs_ddi_out, agg_dd, E_DDI);
        rgcn_aggregate_kernel<<<cdiv((long long)E_PDI * 32, TB), TB, 0, stream>>>(
            hp, pdi_src, pdi_dst, rs_pdi_out, agg_dp, E_PDI);
        rgcn_aggregate_kernel<<<cdiv((long long)E_PPI * 32, TB), TB, 0, stream>>>(
            hp, ppi_src, ppi_dst, rs_ppi_out, agg_pp, E_PPI);

        // drug dst: (agg_dd .* rs_ddi_in) @ W[0] + b[0] + (agg_dp .* rs_pdi_in) @ W[1] + b[1]
        rgcn_gemm_kernel<<<cdiv(ND, 256), TB, 0, stream>>>(
            agg_dd, rs_ddi_in, Wsw + 0 * DIM * DIM, b + 0 * DIM,
            agg_dp, rs_pdi_in, Wsw + 1 * DIM * DIM, b + 1 * DIM,
            hd_next, ND, relu);

        // protein dst: (agg_pp .* rs_ppi_in) @ W[2] + b[2]
        rgcn_gemm_kernel<<<cdiv(NP, 256), TB, 0, stream>>>(
            agg_pp, rs_ppi_in, Wsw + 2 * DIM * DIM, b + 2 * DIM,
            nullptr, nullptr, nullptr, nullptr,
            hp_next, NP, relu);

        hd = hd_next;
        hp = hp_next;
    }
}